// Detector_58179626992027
// MI455X (gfx1250) — compile-verified
//
#include <hip/hip_runtime.h>
#include <hip/hip_bf16.h>
#include <stdint.h>

// YOLOv3 decode for MI455X (gfx1250).
// Memory-bound (116 MB in / 9.5 MB out, ~0.3 FLOP/byte) -> no WMMA content.
// Uses the CDNA5 Tensor Data Mover to DMA a [255 x 64] channel-major tile
// from global into LDS per workgroup (one instruction, TENSORcnt-tracked),
// then computes from LDS and writes coalesced through an LDS staging buffer.

#define CHUNK    64     // cells per block tile (x-dim of TDM tile)
#define NCH      255    // channels = 3 anchors * 85
#define DPC      85     // channels per anchor
#define NTHREADS 192    // 64 cells * 3 anchors, 6 wave32s

typedef __attribute__((ext_vector_type(4))) unsigned int u32x4;
typedef __attribute__((ext_vector_type(8))) int          i32x8;
typedef __attribute__((ext_vector_type(4))) int          i32x4;

#if __has_builtin(__builtin_amdgcn_tensor_load_to_lds) && __has_builtin(__builtin_amdgcn_s_wait_tensorcnt)
#define HAVE_TDM 1
#else
#define HAVE_TDM 0
#endif

__global__ __launch_bounds__(NTHREADS)
void yolo_decode_kernel(const float* __restrict__ in,      // [B, 255, H, W]
                        const float* __restrict__ anchors, // [3, 2]
                        const float* __restrict__ thresh_p,
                        const int*   __restrict__ case_p,
                        float* __restrict__ boxes,         // [Ntot, 6]
                        float* __restrict__ mask,          // [Ntot]
                        int H, int W, float tgrid, long long rowOff)
{
    __shared__ float tile[NCH * CHUNK];        // [channel][cell], 65,280 B
    __shared__ float sbox[CHUNK * 3 * 6];      // staged output rows, 4,608 B

    const int HW  = H * W;
    const int b   = blockIdx.y;
    const int hw0 = blockIdx.x * CHUNK;
    const int t   = threadIdx.x;

    const float* src = in + ((long long)b * NCH) * HW + hw0;

#if HAVE_TDM
    // ---- Tensor Data Mover: one DMA of the [255 x CHUNK] tile into LDS ----
    if (t < 32) {  // wave 0 issues; EXEC is ignored by TDM (one op per wave)
        const uint64_t ga  = (uint64_t)(uintptr_t)src;
        const uint32_t lds = (uint32_t)(uintptr_t)(void*)&tile[0];
        const uint32_t rem = (uint32_t)(HW - hw0);   // clamp x for edge tiles (OOB reads -> 0)

        // D# group 0: count=1 | lds_addr | global_addr[56:0] | type=2
        u32x4 g0;
        g0[0] = 1u;                                   // count=1, no gather
        g0[1] = lds;                                  // lds_addr
        g0[2] = (uint32_t)ga;                         // global_addr[31:0]
        g0[3] = (uint32_t)((ga >> 32) & 0x1FFFFFFu)   // global_addr[56:32]
              | (2u << 30);                           // type=2 ("image")

        // D# group 1
        i32x8 g1;
        g1[0] = 0x00020000;                                  // wg_mask=0, data_size=4B
        g1[1] = (int)((rem & 0xFFFFu) << 16);                // tensor_dim0[15:0] (abar_addr=0)
        g1[2] = (int)((rem >> 16) | ((unsigned)NCH << 16));  // tensor_dim0[31:16] | tensor_dim1[15:0]
        g1[3] = (int)((unsigned)CHUNK << 16);                // tensor_dim1[31:16]=0 | tile_dim0
        g1[4] = NCH;                                         // tile_dim1=255, tile_dim2=0
        g1[5] = HW;                                          // tensor_dim0_stride[31:0]
        g1[6] = 0;                                           // stride hi, dim1_stride lo (unused, 2D)
        g1[7] = 0;
        i32x4 z4 = {0, 0, 0, 0};                             // groups 2/3 unused (2D tensor)
        i32x8 z8 = {0, 0, 0, 0, 0, 0, 0, 0};                 // 6-arg toolchain: extra group

        __builtin_amdgcn_tensor_load_to_lds(g0, g1, z4, z4, z8, 0);
        __builtin_amdgcn_s_wait_tensorcnt(0);
    }
    __syncthreads();
#else
    // ---- Fallback: cooperative coalesced staging of the same tile ----
    for (int i = t; i < NCH * CHUNK; i += NTHREADS) {
        const int c = i / CHUNK, x = i % CHUNK;
        float v = 0.0f;
        if (hw0 + x < HW) v = src[(long long)c * HW + x];
        tile[c * CHUNK + x] = v;
    }
    __syncthreads();
#endif

    // ---- Decode: thread = (anchor a, cell x); LDS reads are bank-conflict-free ----
    const int a  = t / CHUNK;          // 0..2
    const int x  = t % CHUNK;          // 0..63
    const int hw = hw0 + x;
    const bool valid = (hw < HW);

    const float th    = thresh_p[0];
    const float casef = (float)case_p[0];
    const float scale = tgrid / casef;

    const float* ch = &tile[(a * DPC) * CHUNK + x];   // channel c at ch[c*CHUNK]

    const float obj = ch[0];
    const float tx  = ch[1 * CHUNK];
    const float ty  = ch[2 * CHUNK];
    const float tw  = ch[3 * CHUNK];
    const float thh = ch[4 * CHUNK];

    const int gx = hw % W;
    const int gy = hw / W;

    const float p  = 1.0f / (1.0f + __expf(-obj));     // sigmoid
    const float cx = ((float)gx + tx) * scale;
    const float cy = ((float)gy + ty) * scale;
    const float aw = anchors[a * 2 + 0];
    const float ah = anchors[a * 2 + 1];
    const float bw = aw * __expf(tw)  / casef;
    const float bh = ah * __expf(thh) / casef;

    // argmax over 80 class logits; '>' keeps first max (jnp.argmax tie rule)
    float best = ch[5 * CHUNK];
    int   bi   = 0;
#pragma unroll 8
    for (int c = 1; c < DPC - 5; ++c) {
        const float v = ch[(5 + c) * CHUNK];
        if (v > best) { best = v; bi = c; }
    }

    // stage the 6-float row in LDS; local rows are hw-major then anchor,
    // matching the global row order (b*HW+hw)*3 + a -> contiguous per block
    const int rl = x * 3 + a;
    sbox[rl * 6 + 0] = p;
    sbox[rl * 6 + 1] = cx;
    sbox[rl * 6 + 2] = cy;
    sbox[rl * 6 + 3] = bw;
    sbox[rl * 6 + 4] = bh;
    sbox[rl * 6 + 5] = (float)bi;

    if (valid) {
        mask[rowOff + ((long long)b * HW + hw) * 3 + a] = (obj > th) ? 1.0f : 0.0f;
    }
    __syncthreads();

    // ---- Coalesced writeback of the contiguous row block ----
    const int validCells = (HW - hw0 < CHUNK) ? (HW - hw0) : CHUNK;
    const int nFloats    = validCells * 3 * 6;
    const long long base = (rowOff + ((long long)b * HW + hw0) * 3) * 6;
    for (int i = t; i < nFloats; i += NTHREADS) {
        boxes[base + i] = sbox[i];
    }
}

extern "C" void kernel_launch(void* const* d_in, const int* in_sizes, int n_in,
                              void* d_out, int out_size, void* d_ws, size_t ws_size,
                              hipStream_t stream) {
    (void)in_sizes; (void)n_in; (void)out_size; (void)d_ws; (void)ws_size;

    const float* o13 = (const float*)d_in[0];
    const float* o26 = (const float*)d_in[1];
    const float* o52 = (const float*)d_in[2];
    const float* a13 = (const float*)d_in[3];
    const float* a26 = (const float*)d_in[4];
    const float* a52 = (const float*)d_in[5];
    const float* th  = (const float*)d_in[6];
    const int*   cs  = (const int*)  d_in[7];

    const int B = 32;
    const long long N13 = (long long)B * 13 * 13 * 3;   // 16224
    const long long N26 = (long long)B * 26 * 26 * 3;   // 64896
    const long long N52 = (long long)B * 52 * 52 * 3;   // 259584
    const long long NT  = N13 + N26 + N52;              // 340704

    float* boxes = (float*)d_out;
    float* mask  = boxes + NT * 6;

    const dim3 blk(NTHREADS);
    const dim3 g13((13 * 13 + CHUNK - 1) / CHUNK, B);
    const dim3 g26((26 * 26 + CHUNK - 1) / CHUNK, B);
    const dim3 g52((52 * 52 + CHUNK - 1) / CHUNK, B);

    yolo_decode_kernel<<<g13, blk, 0, stream>>>(o13, a13, th, cs, boxes, mask, 13, 13, 32.0f, 0);
    yolo_decode_kernel<<<g26, blk, 0, stream>>>(o26, a26, th, cs, boxes, mask, 26, 26, 16.0f, N13);
    yolo_decode_kernel<<<g52, blk, 0, stream>>>(o52, a52, th, cs, boxes, mask, 52, 52,  8.0f, N13 + N26);
}